// LossCorefLinkerMTT_24790551232724
// MI455X (gfx1250) — compile-verified
//
#include <hip/hip_runtime.h>
#include <hip/hip_bf16.h>
#include <math.h>

typedef __attribute__((ext_vector_type(2))) float v2f;
typedef __attribute__((ext_vector_type(8))) float v8f;

namespace {
constexpr int kBsz = 4;     // batch
constexpr int kN   = 2177;  // full graph size (1 + 128 + 2048)
constexpr int kM   = 2176;  // minor dimension (N-1) = 34 * 64
constexpr int kNB  = 64;    // LU panel width
}

// ---------------------------------------------------------------------------
// 1) Column sums of w = exp(scores)*mask*valid.  One thread per column.
//    m in [0,8): m<4 -> z_mask matrices, m>=4 -> target_mask; b = m&3.
// ---------------------------------------------------------------------------
__global__ void mtt_colsum(const float* __restrict__ scores,
                           const float* __restrict__ tgt,
                           const float* __restrict__ zm,
                           const int*   __restrict__ lengths,
                           float*       __restrict__ cs) {
  const int m = blockIdx.y;
  const int b = m & 3;
  const float* __restrict__ mask = (m < 4) ? zm : tgt;
  const int j = blockIdx.x * blockDim.x + threadIdx.x;
  if (j >= kM) return;
  const int c   = j + 1;
  const int len = lengths[b];
  float s = 0.0f;
  if (c < len) {
    const size_t base = (size_t)b * kN * kN + (size_t)c;
    for (int r = 0; r < len; ++r) {           // valid(r) via loop bound
      const size_t idx = base + (size_t)r * kN;
      s += expf(scores[idx]) * mask[idx];
    }
  }
  cs[(size_t)m * kM + j] = s;
}

// ---------------------------------------------------------------------------
// 2) Materialize the Laplacian minor:  A = diag(colsum + pad) - w
// ---------------------------------------------------------------------------
__global__ void mtt_build(const float* __restrict__ scores,
                          const float* __restrict__ tgt,
                          const float* __restrict__ zm,
                          const int*   __restrict__ lengths,
                          const float* __restrict__ cs,
                          float*       __restrict__ Amat) {
  const int m = blockIdx.z;
  const int b = m & 3;
  const float* __restrict__ mask = (m < 4) ? zm : tgt;
  const int j = blockIdx.x * 16 + threadIdx.x;   // col of minor
  const int i = blockIdx.y * 16 + threadIdx.y;   // row of minor
  const int c = j + 1, r = i + 1;
  const int len = lengths[b];
  float w = 0.0f;
  if (r < len && c < len) {
    const size_t idx = (size_t)b * kN * kN + (size_t)r * kN + c;
    w = expf(scores[idx]) * mask[idx];
  }
  float v = -w;
  if (i == j) v += cs[(size_t)m * kM + j] + ((c >= len) ? 1.0f : 0.0f);
  Amat[(size_t)m * kM * kM + (size_t)i * kM + j] = v;
}

// ---------------------------------------------------------------------------
// 3a) Panel factorization (no pivoting; minor is column diagonally dominant).
//     One 256-thread workgroup per matrix factors A[k:M, k:k+64].
//     NOTE: sub-diagonal L entries are stored NEGATED (-L), so the Schur
//     update needs no per-fragment negation (f32 WMMA NEG only touches C).
//     L is dead after the update; only pivots feed the logdet.
// ---------------------------------------------------------------------------
__global__ void mtt_panel(float* __restrict__ wsA, int k,
                          float* __restrict__ ld) {
  const int m = blockIdx.x;
  float* __restrict__ A = wsA + (size_t)m * kM * kM;
  __shared__ float s_prow[kNB];
  const int tid = threadIdx.x;
  float acc = 0.0f;
  for (int j = 0; j < kNB; ++j) {
    const int col = k + j;
    __syncthreads();                                  // prior updates visible
    if (tid < kNB) s_prow[tid] = A[(size_t)col * kM + k + tid];
    __syncthreads();
    const float piv = s_prow[j];
    const float rp  = 1.0f / piv;
    for (int i = col + 1 + tid; i < kM; i += blockDim.x) {
      const size_t ro  = (size_t)i * kM;
      const float  lij = A[ro + col] * rp;            // true L value
      A[ro + col] = -lij;                             // store -L
      for (int jj = j + 1; jj < kNB; ++jj)
        A[ro + k + jj] -= lij * s_prow[jj];
    }
    if (tid == 0) acc += logf(fabsf(piv));
  }
  if (tid == 0) ld[m] += acc;                         // single writer per m
}

// ---------------------------------------------------------------------------
// 3b) Row-panel TRSM:  U12 = L11^{-1} * A12.  L11 is stored as -L below the
//     diagonal, so forward substitution becomes a[i] += sum (-L)[i][j]*a[j].
// ---------------------------------------------------------------------------
__global__ void mtt_trsm(float* __restrict__ wsA, int k) {
  const int m = blockIdx.y;
  float* __restrict__ A = wsA + (size_t)m * kM * kM;
  __shared__ float sL[kNB][kNB];
  const int tid = threadIdx.x;
  for (int idx = tid; idx < kNB * kNB; idx += blockDim.x) {
    const int i = idx >> 6, jj = idx & 63;
    sL[i][jj] = A[(size_t)(k + i) * kM + k + jj];
  }
  __syncthreads();
  const int c = k + kNB + blockIdx.x * blockDim.x + tid;
  if (c >= kM) return;
  float a[kNB];
  for (int i = 0; i < kNB; ++i) a[i] = A[(size_t)(k + i) * kM + c];
  for (int i = 1; i < kNB; ++i) {
    float s = a[i];
    for (int jj = 0; jj < i; ++jj) s += sL[i][jj] * a[jj];   // sL = -L
    a[i] = s;
  }
  for (int i = 0; i < kNB; ++i) A[(size_t)(k + i) * kM + c] = a[i];
}

// ---------------------------------------------------------------------------
// 3c) Schur update  A22 += (-L21) * U12   via f32 WMMA (16x16x4).
//     128 threads (4 waves) per 64x64 output tile; waves arranged 2x2, each
//     computes a 32x32 quadrant as 2x2 grid of 16x16 f32 accumulators.
//     sA padded to 65 floats/row: column-direction reads hit distinct banks.
// ---------------------------------------------------------------------------
__global__ void __launch_bounds__(128) mtt_gemm(float* __restrict__ wsA,
                                                int k) {
  const int m = blockIdx.z;
  float* __restrict__ A = wsA + (size_t)m * kM * kM;
  const int base = k + kNB;
  const int row0 = base + blockIdx.y * 64;
  const int col0 = base + blockIdx.x * 64;
  const int tid  = threadIdx.x;
  const int lane = tid & 31;
  const int wave = tid >> 5;
  const int wr   = (wave >> 1) * 32;   // wave's row quadrant
  const int wc   = (wave & 1) * 32;    // wave's col quadrant

  __shared__ float sA[64][65];   // -L21 tile (rows row0.., cols k..k+63)
  __shared__ float sB[64][64];   // U12 tile (rows k..k+63, cols col0..)
  {
    const int kk = tid & 63;
    const int i0 = tid >> 6;                 // 0 or 1
    for (int it = 0; it < 32; ++it) {
      const int i = i0 + 2 * it;
      sA[i][kk] = A[(size_t)(row0 + i) * kM + (k + kk)];
      sB[i][kk] = A[(size_t)(k + i)   * kM + (col0 + kk)];
    }
  }
  __syncthreads();

  const int n  = lane & 15;      // N index within 16x16 tile / A-frag row
  const int hi = lane >> 4;      // half-wave select

  // C/D layout (ISA 7.12.2): VGPR v, lane L -> M = v + 8*hi, N = L%16
  v8f c[2][2];
  for (int a = 0; a < 2; ++a)
    for (int b2 = 0; b2 < 2; ++b2)
      for (int v = 0; v < 8; ++v)
        c[a][b2][v] = A[(size_t)(row0 + wr + 16 * a + v + 8 * hi) * kM +
                        (col0 + wc + 16 * b2 + n)];

  for (int kk = 0; kk < kNB; kk += 4) {
    const int ks = kk + 2 * hi;  // A 16x4 f32: lanes 0-15 K={0,1}, 16-31 K={2,3}
    v2f af[2], bf[2];
    for (int a = 0; a < 2; ++a) {
      af[a].x = sA[wr + 16 * a + n][ks];       // already -L
      af[a].y = sA[wr + 16 * a + n][ks + 1];
    }
    for (int b2 = 0; b2 < 2; ++b2) {
      bf[b2].x = sB[ks][wc + 16 * b2 + n];
      bf[b2].y = sB[ks + 1][wc + 16 * b2 + n];
    }
    for (int a = 0; a < 2; ++a)
      for (int b2 = 0; b2 < 2; ++b2)
        c[a][b2] = __builtin_amdgcn_wmma_f32_16x16x4_f32(
            false, af[a], false, bf[b2], (short)0, c[a][b2], false, false);
  }

  for (int a = 0; a < 2; ++a)
    for (int b2 = 0; b2 < 2; ++b2)
      for (int v = 0; v < 8; ++v)
        A[(size_t)(row0 + wr + 16 * a + v + 8 * hi) * kM +
          (col0 + wc + 16 * b2 + n)] = c[a][b2][v];
}

// ---------------------------------------------------------------------------
// 4) init / finalize
// ---------------------------------------------------------------------------
__global__ void mtt_init(float* __restrict__ ld) {
  if (threadIdx.x < 8) ld[threadIdx.x] = 0.0f;
}

__global__ void mtt_final(const float* __restrict__ ld,
                          float* __restrict__ out) {
  float s = 0.0f;
  for (int b = 0; b < kBsz; ++b) s += ld[b] - ld[4 + b];
  out[0] = s * 0.25f;
}

// ---------------------------------------------------------------------------
extern "C" void kernel_launch(void* const* d_in, const int* in_sizes, int n_in,
                              void* d_out, int out_size, void* d_ws,
                              size_t ws_size, hipStream_t stream) {
  (void)in_sizes; (void)n_in; (void)out_size; (void)ws_size;
  const float* scores  = (const float*)d_in[0];
  const float* tgt     = (const float*)d_in[1];
  const float* zm      = (const float*)d_in[2];
  const int*   lengths = (const int*)d_in[3];

  // workspace layout: 8 matrices | 8 colsum vectors | 8 logdet accumulators
  float* ws   = (float*)d_ws;
  const size_t As = (size_t)kM * kM;
  float* Amat = ws;
  float* cs   = ws + 8 * As;
  float* ld   = cs + (size_t)8 * kM;

  mtt_init<<<1, 32, 0, stream>>>(ld);
  mtt_colsum<<<dim3((kM + 255) / 256, 8), 256, 0, stream>>>(scores, tgt, zm,
                                                            lengths, cs);
  mtt_build<<<dim3(kM / 16, kM / 16, 8), dim3(16, 16), 0, stream>>>(
      scores, tgt, zm, lengths, cs, Amat);

  for (int k = 0; k < kM; k += kNB) {
    mtt_panel<<<8, 256, 0, stream>>>(Amat, k, ld);
    const int rem = kM - k - kNB;
    if (rem > 0) {
      mtt_trsm<<<dim3((rem + 255) / 256, 8), 256, 0, stream>>>(Amat, k);
      mtt_gemm<<<dim3(rem / 64, rem / 64, 8), 128, 0, stream>>>(Amat, k);
    }
  }

  mtt_final<<<1, 1, 0, stream>>>(ld, (float*)d_out);
}